// ClusterScaleBiasBlock_54915451847279
// MI455X (gfx1250) — compile-verified
//
#include <hip/hip_runtime.h>

// Problem constants from the reference.
#define F_DIM 2048
#define C_DIM 64
#define BN_EPS 1e-3f

typedef float v4f __attribute__((ext_vector_type(4)));

// ---------------------------------------------------------------------------
// HOT KERNEL (placed first so the disasm snippet shows it).
//
// out[row,f] = (x*scale + shift) * (1 + Wg[cid,f]) + Wb[cid,f]
//
// One block = 2 rows. 256 threads (8 wave32) x 2 float4 x 2 rows.
// - cluster ids are uniform per block -> readfirstlane -> SGPR row bases.
// - x / out are pure streaming (no reuse): non-temporal b128 ops so the
//   512 MB stream does not evict the 1 MB Wg/Wb working set (L2-resident,
//   re-read ~512x) nor the 16 KB scale/shift vectors.
// - scale/shift loaded once per float4-column, reused for both rows
//   (cuts cached traffic 16B/elem -> 12B/elem).
// ---------------------------------------------------------------------------
__global__ __launch_bounds__(256) void cluster_scale_bias_kernel(
    const float* __restrict__ x,
    const float* __restrict__ Wg,
    const float* __restrict__ Wb,
    const int*   __restrict__ cluster_ids,
    const float* __restrict__ scale,
    const float* __restrict__ shift,
    float* __restrict__ out)
{
    const int row0 = blockIdx.x * 2;
    // Uniform per block: force scalar so row bases live in SGPRs.
    const int cid0 = __builtin_amdgcn_readfirstlane(cluster_ids[row0]);
    const int cid1 = __builtin_amdgcn_readfirstlane(cluster_ids[row0 + 1]);

    const size_t r0_off = (size_t)row0 * F_DIM;
    const size_t r1_off = r0_off + F_DIM;
    const size_t c0_off = (size_t)cid0 * F_DIM;
    const size_t c1_off = (size_t)cid1 * F_DIM;

    const v4f* __restrict__ xr0 = (const v4f*)(x  + r0_off);
    const v4f* __restrict__ xr1 = (const v4f*)(x  + r1_off);
    const v4f* __restrict__ gr0 = (const v4f*)(Wg + c0_off);
    const v4f* __restrict__ gr1 = (const v4f*)(Wg + c1_off);
    const v4f* __restrict__ br0 = (const v4f*)(Wb + c0_off);
    const v4f* __restrict__ br1 = (const v4f*)(Wb + c1_off);
    const v4f* __restrict__ sc  = (const v4f*)scale;
    const v4f* __restrict__ sh  = (const v4f*)shift;
    v4f* __restrict__ o0        = (v4f*)(out + r0_off);
    v4f* __restrict__ o1        = (v4f*)(out + r1_off);

#pragma unroll
    for (int i = 0; i < 2; ++i) {
        const int idx = (int)threadIdx.x + i * 256;   // 0..511 float4 per row

        // Shared-per-column BN affine (cached, reused across both rows).
        v4f sv = sc[idx];
        v4f tv = sh[idx];

        // Streaming inputs: non-temporal 128-bit loads.
        v4f x0 = __builtin_nontemporal_load(xr0 + idx);
        v4f x1 = __builtin_nontemporal_load(xr1 + idx);

        // Gathered per-cluster rows (L2-resident, regular temporal loads).
        v4f g0 = gr0[idx];
        v4f b0 = br0[idx];
        v4f g1 = gr1[idx];
        v4f b1 = br1[idx];

        // (x*scale + shift) * (1 + g) + b   -> packed FMAs
        v4f r0 = (x0 * sv + tv) * (g0 + 1.0f) + b0;
        v4f r1 = (x1 * sv + tv) * (g1 + 1.0f) + b1;

        // Streaming outputs: non-temporal 128-bit stores.
        __builtin_nontemporal_store(r0, o0 + idx);
        __builtin_nontemporal_store(r1, o1 + idx);
    }
}

// ---------------------------------------------------------------------------
// Kernel 2: fold BatchNorm inference params into scale/shift (F elements).
//   scale = gamma * rsqrt(var + eps)
//   shift = beta  - mean * scale
// 2048 elements -> negligible cost; done precisely with sqrtf.
// ---------------------------------------------------------------------------
__global__ __launch_bounds__(256) void bn_fold_kernel(
    const float* __restrict__ bn_gamma,
    const float* __restrict__ bn_beta,
    const float* __restrict__ moving_mean,
    const float* __restrict__ moving_var,
    float* __restrict__ scale,
    float* __restrict__ shift)
{
    const int f = blockIdx.x * blockDim.x + threadIdx.x;
    if (f < F_DIM) {
        const float s = bn_gamma[f] / sqrtf(moving_var[f] + BN_EPS);
        scale[f] = s;
        shift[f] = bn_beta[f] - moving_mean[f] * s;
    }
}

// ---------------------------------------------------------------------------
// Launch wrapper. Input order (setup_inputs dict order):
//   0: x [B,F] f32      1: Wg [C,F] f32    2: Wb [C,F] f32
//   3: bn_gamma [F]     4: bn_beta [F]     5: moving_mean [F]
//   6: moving_var [F]   7: cluster_ids [B] i32
// d_ws: scale[F] | shift[F]  (16 KB)
// ---------------------------------------------------------------------------
extern "C" void kernel_launch(void* const* d_in, const int* in_sizes, int n_in,
                              void* d_out, int out_size, void* d_ws, size_t ws_size,
                              hipStream_t stream)
{
    const float* x           = (const float*)d_in[0];
    const float* Wg          = (const float*)d_in[1];
    const float* Wb          = (const float*)d_in[2];
    const float* bn_gamma    = (const float*)d_in[3];
    const float* bn_beta     = (const float*)d_in[4];
    const float* moving_mean = (const float*)d_in[5];
    const float* moving_var  = (const float*)d_in[6];
    const int*   cluster_ids = (const int*)d_in[7];
    float*       out         = (float*)d_out;

    float* scale = (float*)d_ws;
    float* shift = scale + F_DIM;

    const int B = in_sizes[0] / F_DIM;   // 32768

    bn_fold_kernel<<<(F_DIM + 255) / 256, 256, 0, stream>>>(
        bn_gamma, bn_beta, moving_mean, moving_var, scale, shift);

    cluster_scale_bias_kernel<<<B / 2, 256, 0, stream>>>(
        x, Wg, Wb, cluster_ids, scale, shift, out);
}